// Laplacian3D_39264591020047
// MI455X (gfx1250) — compile-verified
//
#include <hip/hip_runtime.h>
#include <hip/hip_bf16.h>

// ---------------------------------------------------------------------------
// 3D Laplacian (7-point stencil, zero padding) on (8,1,192,192,192) f32.
// Memory-bound: ~453 MB min traffic -> ~19.5 us floor at 23.3 TB/s HBM.
// Strategy: async global->LDS staging (CDNA5 GLOBAL_LOAD_ASYNC_TO_LDS_B128,
// ASYNCcnt + s_wait_asynccnt) with a 4-buffer sliding window of D-planes,
// full-width W rows (no W halo), TH=16 H-rows per tile, prefetch of plane
// d+2 overlapped with compute of plane d.
// ---------------------------------------------------------------------------

#define kN   8
#define kD   192
#define kH   192
#define kW   192
#define TH   16                 // H rows computed per block per D-plane
#define DCH  16                 // D planes per block (chunk)
#define ROWS (TH + 2)           // 18 rows incl. halo
#define PLANE (ROWS * kW)       // 3456 floats = 13824 B, contiguous in memory
#define CHUNKS (PLANE / 4)      // 864 float4 chunks per plane
#define NTHREADS 192            // 6 wave32; one thread per W column
#define NSLOT 5                 // chunk slots per thread: 4 full + 1 tail

// stencil coefficients (DX=DY=DZ=0.01)
#define CXD 10000.0f            // D axis: 1/DX^2
#define CYH 10000.0f            // H axis: 1/DY^2
#define CZW 10000.0f            // W axis: 1/DZ^2
#define C0  (-2.0f * (CXD + CYH + CZW))

typedef int v4i __attribute__((ext_vector_type(4)));

__device__ __forceinline__ void async_copy_b128(const float* g, float* l) {
#if __has_builtin(__builtin_amdgcn_global_load_async_to_lds_b128)
  // arg0 = global (AS1) src, arg1 = LDS dst, imm offset, imm cpol.
  __builtin_amdgcn_global_load_async_to_lds_b128((v4i*)g, (v4i*)l,
                                                 /*offset=*/0, /*cpol=*/0);
#else
  unsigned lds_addr = (unsigned)(unsigned long long)l;   // LDS byte addr = addr[31:0]
  asm volatile("global_load_async_to_lds_b128 %0, %1, off"
               :: "v"(lds_addr), "v"((unsigned long long)g)
               : "memory");
#endif
}

__device__ __forceinline__ void wait_async_zero() {
#if __has_builtin(__builtin_amdgcn_s_wait_asynccnt)
  __builtin_amdgcn_s_wait_asynccnt(0);
#else
  asm volatile("s_wait_asynccnt 0" ::: "memory");
#endif
}

__global__ __launch_bounds__(NTHREADS)
void lap3d_kernel(const float* __restrict__ x, float* __restrict__ out) {
  __shared__ float buf[4][PLANE];

  const int t  = threadIdx.x;           // also the W column this thread owns
  const int h0 = blockIdx.x * TH;       // first computed H row
  const int d0 = blockIdx.y * DCH;      // first computed D plane
  const int n  = blockIdx.z;

  const float* xn = x   + (size_t)n * (kD * kH * kW);
  float*       on = out + (size_t)n * (kD * kH * kW);

  // Per-thread chunk geometry is invariant across planes: hoist it.
  // Chunk ids: c = t + 192*k (k=0..3), plus tail c = 768 + t for t < 96.
  int  off[NSLOT];      // float offset of chunk within a D-plane: h*kW + col*4
  int  lofs[NSLOT];     // float offset within the LDS plane buffer: c*4
  bool hok[NSLOT];      // row in [0, kH)
  bool act[NSLOT];      // slot active for this thread
#pragma unroll
  for (int k = 0; k < NSLOT; ++k) {
    const int c    = (k < 4) ? (t + NTHREADS * k) : (4 * NTHREADS + t);
    act[k]  = (k < 4) || (t < CHUNKS - 4 * NTHREADS);
    const int r    = c / (kW / 4);
    const int col4 = c % (kW / 4);
    const int h    = h0 - 1 + r;
    hok[k]  = (h >= 0) && (h < kH);
    off[k]  = h * kW + col4 * 4;        // only used when hok[k]
    lofs[k] = c * 4;
  }

  // Stage plane `d` (rows [h0-1, h0+TH]) into buf[b]; zero-fill out-of-range
  // rows/planes (zero-padding semantics). Rows of one D-plane are contiguous
  // (row stride == row length), so the async b128 loads are fully coalesced.
  auto load_plane = [&](int d, int b) {
    float* lp = &buf[b][0];
    const bool d_ok = (d >= 0) && (d < kD);
    const float* gp = xn + (size_t)(d_ok ? d : 0) * (kH * kW);  // in-bounds math
    const float4 z4 = {0.f, 0.f, 0.f, 0.f};
#pragma unroll
    for (int k = 0; k < NSLOT; ++k) {
      if (act[k]) {
        if (d_ok && hok[k])
          async_copy_b128(gp + off[k], lp + lofs[k]);
        else
          *(float4*)(lp + lofs[k]) = z4;
      }
    }
  };

  // Prologue: planes d0-1, d0, d0+1 -> buffers 0,1,2. Plane p lives in
  // buffer ((p - d0) + 1) & 3.
  load_plane(d0 - 1, 0);
  load_plane(d0,     1);
  load_plane(d0 + 1, 2);
  wait_async_zero();
  __syncthreads();

  const int w = t;
  for (int i = 0; i < DCH; ++i) {
    const int d = d0 + i;
    // Prefetch plane d+2 into the buffer whose last reader finished at the
    // previous barrier; overlaps with this iteration's compute.
    if (i + 1 < DCH) load_plane(d + 2, (i + 3) & 3);

    const float* prv = buf[i & 3];        // plane d-1
    const float* cur = buf[(i + 1) & 3];  // plane d
    const float* nxt = buf[(i + 2) & 3];  // plane d+1

    float below  = cur[w];                // local row 0 (h0-1 halo)
    float center = cur[kW + w];           // local row 1 (h0)
    float* orow = on + ((size_t)d * kH + h0) * kW + w;
#pragma unroll
    for (int r = 1; r <= TH; ++r) {       // global h = h0 + r - 1
      const float above = cur[(r + 1) * kW + w];
      const float left  = (w == 0)      ? 0.f : cur[r * kW + w - 1];
      const float right = (w == kW - 1) ? 0.f : cur[r * kW + w + 1];
      const float pz    = prv[r * kW + w];
      const float nz    = nxt[r * kW + w];
      const float v = C0  * center
                    + CYH * (below + above)
                    + CZW * (left + right)
                    + CXD * (pz + nz);
      orow[(size_t)(r - 1) * kW] = v;     // coalesced across the 192 lanes
      below  = center;
      center = above;
    }

    wait_async_zero();   // prefetched plane resident in LDS for this wave
    __syncthreads();     // visible to all waves; safe to recycle buffers
  }
}

extern "C" void kernel_launch(void* const* d_in, const int* in_sizes, int n_in,
                              void* d_out, int out_size, void* d_ws, size_t ws_size,
                              hipStream_t stream) {
  const float* x = (const float*)d_in[0];
  float* out = (float*)d_out;
  dim3 grid(kH / TH, kD / DCH, kN);   // (12, 12, 8) = 1152 blocks
  lap3d_kernel<<<grid, NTHREADS, 0, stream>>>(x, out);
}